// ProbAttention_90941637525896
// MI455X (gfx1250) — compile-verified
//
#include <hip/hip_runtime.h>
#include <hip/hip_bf16.h>
#include <math.h>

// Problem constants (from reference): B=8, L=2048, H=8, D=64, u = 5*ceil(ln 2048) = 40
#define BB   8
#define LL   2048
#define HH   8
#define DD   64
#define UU   40
#define UPAD 48            // 40 padded to 3 WMMA M-tiles of 16
#define NBH  (BB * HH)     // 64 (b,h) pairs
#define SCALE 0.125f       // 1/sqrt(64)

typedef __attribute__((ext_vector_type(2))) float v2f;
typedef __attribute__((ext_vector_type(8))) float v8f;

__device__ __forceinline__ v8f wmma_f32(v2f a, v2f b, v8f c) {
  // V_WMMA_F32_16X16X4_F32: D = A(16x4) x B(4x16) + C(16x16), fp32 throughout
  return __builtin_amdgcn_wmma_f32_16x16x4_f32(false, a, false, b, (short)0, c,
                                               false, false);
}

__device__ __forceinline__ v8f v8f_zero() {
  v8f z;
#pragma unroll
  for (int i = 0; i < 8; ++i) z[i] = 0.0f;
  return z;
}

// ---------------------------------------------------------------------------
// Phase 1: M[b,h,l] = max_s(q.k_idx) - sum_s(q.k_idx)/L   (one thread per row)
// ---------------------------------------------------------------------------
__global__ void k_sample_scores(const float* __restrict__ q,
                                const float* __restrict__ kmat,
                                const int* __restrict__ idx,
                                float* __restrict__ Mout) {
  const int gid = blockIdx.x * blockDim.x + threadIdx.x;
  if (gid >= NBH * LL) return;
  const int bh = gid / LL;
  const int l  = gid % LL;

  const float4* qrow = (const float4*)(q + ((size_t)bh * LL + l) * DD);
  float4 qr[DD / 4];
#pragma unroll
  for (int i = 0; i < DD / 4; ++i) qr[i] = qrow[i];

  const int* irow = idx + (size_t)l * UU;
  float mx = -__builtin_inff();
  float sm = 0.0f;
  for (int s = 0; s < UU; ++s) {
    const int ki = irow[s];
    const float4* krow = (const float4*)(kmat + ((size_t)bh * LL + ki) * DD);
    float acc = 0.0f;
#pragma unroll
    for (int i = 0; i < DD / 4; ++i) {
      const float4 kv = krow[i];
      acc = fmaf(qr[i].x, kv.x, acc);
      acc = fmaf(qr[i].y, kv.y, acc);
      acc = fmaf(qr[i].z, kv.z, acc);
      acc = fmaf(qr[i].w, kv.w, acc);
    }
    mx = fmaxf(mx, acc);
    sm += acc;
  }
  Mout[gid] = mx - sm * (1.0f / (float)LL);
}

// ---------------------------------------------------------------------------
// Phase 2: per (b,h) top-40 of M (stable ties -> lower index, like lax.top_k),
//          then gather Q_reduce into zero-padded 48x64 tile in workspace.
// ---------------------------------------------------------------------------
__global__ void k_topk_gather(const float* __restrict__ Mws,
                              const float* __restrict__ q,
                              float* __restrict__ Qr) {
  const int bh = blockIdx.x;
  const int t  = threadIdx.x;           // 256 threads
  __shared__ float sM[LL];              // 8 KB
  __shared__ float rv[256];
  __shared__ int   ri[256];
  __shared__ int   topIdx[UU];

  for (int i = t; i < LL; i += 256) sM[i] = Mws[(size_t)bh * LL + i];
  __syncthreads();

  for (int k = 0; k < UU; ++k) {
    float bv = -__builtin_inff();
    int   bi = 0x7fffffff;
    for (int i = t; i < LL; i += 256) {   // increasing index: strict > keeps lowest
      const float v = sM[i];
      if (v > bv) { bv = v; bi = i; }
    }
    rv[t] = bv; ri[t] = bi;
    __syncthreads();
    for (int off = 128; off > 0; off >>= 1) {
      if (t < off) {
        const float ov = rv[t + off];
        const int   oi = ri[t + off];
        if (ov > rv[t] || (ov == rv[t] && oi < ri[t])) { rv[t] = ov; ri[t] = oi; }
      }
      __syncthreads();
    }
    if (t == 0) { topIdx[k] = ri[0]; sM[ri[0]] = -__builtin_inff(); }
    __syncthreads();
  }

  // Gather Q_reduce (UPAD x DD), pad rows are zero.
  for (int e = t; e < UPAD * DD; e += 256) {
    const int r = e / DD;
    const int d = e % DD;
    float val = 0.0f;
    if (r < UU) val = q[((size_t)bh * LL + topIdx[r]) * DD + d];
    Qr[((size_t)bh * UPAD + r) * DD + e % DD] = val;
  }
}

// ---------------------------------------------------------------------------
// Phase 3a: scores = Q_reduce(48x64) @ k^T(64x2048) via V_WMMA_F32_16X16X4_F32.
// Block = 4 waves; wave w owns N-tile (blockIdx.y*4+w); B-tile loaded once per
// K-step and reused across the 3 M-tiles (3 wmma per B load).
// ---------------------------------------------------------------------------
__global__ void k_scores(const float* __restrict__ Qr,
                         const float* __restrict__ kmat,
                         float* __restrict__ attn) {
  __shared__ float sQ[UPAD * DD];       // 12 KB
  const int bh   = blockIdx.x;
  const int lane = threadIdx.x & 31;
  const int wave = threadIdx.x >> 5;

  for (int i = threadIdx.x; i < UPAD * DD; i += blockDim.x)
    sQ[i] = Qr[(size_t)bh * UPAD * DD + i];
  __syncthreads();

  const int n0      = (blockIdx.y * 4 + wave) * 16;
  const int halfsel = lane >> 4;        // 0: lanes 0-15, 1: lanes 16-31
  const int lsub    = lane & 15;
  const int koff    = halfsel * 2;      // ISA A/B layout: upper half holds K+2,K+3
  const float* kbase = kmat + (size_t)bh * LL * DD;

  v8f acc[3];
#pragma unroll
  for (int mt = 0; mt < 3; ++mt) acc[mt] = v8f_zero();

#pragma unroll
  for (int kk = 0; kk < DD; kk += 4) {
    const int kb = kk + koff;
    const float* kp = kbase + (size_t)(n0 + lsub) * DD + kb;  // B[k][n] = k[n][k]
    v2f b; b.x = kp[0]; b.y = kp[1];
#pragma unroll
    for (int mt = 0; mt < 3; ++mt) {
      const float* ap = &sQ[(mt * 16 + lsub) * DD + kb];
      v2f a; a.x = ap[0]; a.y = ap[1];
      acc[mt] = wmma_f32(a, b, acc[mt]);
    }
  }

  // C layout: VGPR r -> M = r (lanes 0-15) / M = r+8 (lanes 16-31), N = lane&15
#pragma unroll
  for (int mt = 0; mt < 3; ++mt) {
#pragma unroll
    for (int r = 0; r < 8; ++r) {
      const int row = mt * 16 + r + halfsel * 8;
      if (row < UU)
        attn[((size_t)bh * UU + row) * LL + n0 + lsub] = acc[mt][r] * SCALE;
    }
  }
}

// ---------------------------------------------------------------------------
// Phase 3b: in-place row softmax over L=2048 (one block per (b,h,u) row).
// ---------------------------------------------------------------------------
__global__ void k_softmax(float* __restrict__ attn) {
  const int row = blockIdx.x;           // B*H*U rows
  float* p = attn + (size_t)row * LL;
  __shared__ float red[256];
  const int t = threadIdx.x;

  float mx = -__builtin_inff();
  for (int i = t; i < LL; i += 256) mx = fmaxf(mx, p[i]);
  red[t] = mx; __syncthreads();
  for (int off = 128; off > 0; off >>= 1) {
    if (t < off) red[t] = fmaxf(red[t], red[t + off]);
    __syncthreads();
  }
  mx = red[0];
  __syncthreads();

  float sm = 0.0f;
  for (int i = t; i < LL; i += 256) {
    const float e = __expf(p[i] - mx);
    p[i] = e;
    sm += e;
  }
  red[t] = sm; __syncthreads();
  for (int off = 128; off > 0; off >>= 1) {
    if (t < off) red[t] += red[t + off];
    __syncthreads();
  }
  const float inv = 1.0f / red[0];
  for (int i = t; i < LL; i += 256) p[i] *= inv;
}

// ---------------------------------------------------------------------------
// Phase 3c: context = attn(40x2048) @ v(2048x64) via V_WMMA_F32_16X16X4_F32.
// Block = 4 waves (one 16-wide N-tile each); 512 K-steps, 3 wmma per B load.
// ---------------------------------------------------------------------------
__global__ void k_context(const float* __restrict__ attn,
                          const float* __restrict__ vmat,
                          float* __restrict__ ctx) {
  const int bh   = blockIdx.x;
  const int lane = threadIdx.x & 31;
  const int wave = threadIdx.x >> 5;
  const int n0      = wave * 16;
  const int halfsel = lane >> 4;
  const int lsub    = lane & 15;
  const int koff    = halfsel * 2;

  const float* abase = attn + (size_t)bh * UU * LL;
  const float* vbase = vmat + (size_t)bh * LL * DD;

  v8f acc[3];
#pragma unroll
  for (int mt = 0; mt < 3; ++mt) acc[mt] = v8f_zero();

  for (int kk = 0; kk < LL; kk += 4) {
    const int kb = kk + koff;
    v2f b;                                         // B[k][n] = v[k][n] (natural)
    b.x = vbase[(size_t)(kb)     * DD + n0 + lsub];
    b.y = vbase[(size_t)(kb + 1) * DD + n0 + lsub];
#pragma unroll
    for (int mt = 0; mt < 3; ++mt) {
      const int row = mt * 16 + lsub;
      v2f a;
      if (row < UU) {                              // pad rows feed zeros (no OOB)
        const float* ap = abase + (size_t)row * LL + kb;
        a.x = ap[0]; a.y = ap[1];
      } else {
        a.x = 0.0f; a.y = 0.0f;
      }
      acc[mt] = wmma_f32(a, b, acc[mt]);
    }
  }

#pragma unroll
  for (int mt = 0; mt < 3; ++mt) {
#pragma unroll
    for (int r = 0; r < 8; ++r) {
      const int row = mt * 16 + r + halfsel * 8;
      if (row < UU)
        ctx[((size_t)bh * UU + row) * DD + n0 + lsub] = acc[mt][r];
    }
  }
}

// ---------------------------------------------------------------------------
extern "C" void kernel_launch(void* const* d_in, const int* in_sizes, int n_in,
                              void* d_out, int out_size, void* d_ws, size_t ws_size,
                              hipStream_t stream) {
  const float* q    = (const float*)d_in[0];  // (B,L,H,D) flat == (B,H,L,D) reshape
  const float* kmat = (const float*)d_in[1];
  const float* vmat = (const float*)d_in[2];
  const int*   idx  = (const int*)d_in[3];    // (L, U)

  float* out     = (float*)d_out;
  float* context = out;                       // B*H*U*D floats
  float* attn    = out + (size_t)NBH * UU * DD;  // B*H*U*L floats

  float* Mws = (float*)d_ws;                  // B*H*L floats      (512 KB)
  float* Qr  = Mws + (size_t)NBH * LL;        // B*H*48*64 floats  (768 KB)

  // Phase 1: sparsity measure M
  {
    const int total = NBH * LL;
    k_sample_scores<<<(total + 255) / 256, 256, 0, stream>>>(q, kmat, idx, Mws);
  }
  // Phase 2: top-k + Q_reduce gather
  k_topk_gather<<<NBH, 256, 0, stream>>>(Mws, q, Qr);
  // Phase 3a: scores GEMM (WMMA f32), scaled, into attn region
  k_scores<<<dim3(NBH, LL / 64), 128, 0, stream>>>(Qr, kmat, attn);
  // Phase 3b: softmax rows in place
  k_softmax<<<NBH * UU, 256, 0, stream>>>(attn);
  // Phase 3c: context GEMM (WMMA f32)
  k_context<<<NBH, 128, 0, stream>>>(attn, vmat, context);
}